// LSTMTagger_CRF_70712341561597
// MI455X (gfx1250) — compile-verified
//
#include <hip/hip_runtime.h>

// Viterbi decode (CRF) for B x T x K, T=K=128, on gfx1250.
// One workgroup per batch element; 128 threads (4 wave32) per workgroup.
// CDNA5 features: TDM tensor_load_to_lds (async 64KB feature tile -> LDS),
// s_wait_tensorcnt, 320KB-per-WGP LDS residency, conflict-free ds_load_b128
// via padded-transpose transition matrix, ping-pong partition (1 barrier/step).

#define T_LEN 128
#define K_TAGS 128
#define PITCH 132           // 128 + 4 floats: bank = (4c + prev) % 64, conflict-free
#define START_TAG (K_TAGS - 2)
#define STOP_TAG  (K_TAGS - 1)

typedef unsigned int v4u __attribute__((ext_vector_type(4)));
typedef int          v4i __attribute__((ext_vector_type(4)));
typedef int          v8i __attribute__((ext_vector_type(8)));

__global__ __launch_bounds__(128) void viterbi_crf_kernel(
    const float* __restrict__ feats,          // (B,T,K)
    const float* __restrict__ trans,          // (K,K)  trans[prev][cur]
    const unsigned char* __restrict__ masks,  // (B,T)  bool
    float* __restrict__ out_score,            // (B,)
    int* __restrict__ out_idx)                // (B,T)
{
    __shared__ __align__(16) float feats_lds[T_LEN * K_TAGS];   // 64 KB (TDM target)
    __shared__ __align__(16) float transT[K_TAGS * PITCH];      // 66 KB transposed+padded
    __shared__ __align__(16) float part_ab[2][K_TAGS];          // ping-pong partition
    __shared__ __align__(16) float part_save[K_TAGS];
    __shared__ __align__(16) float red_f[K_TAGS];
    __shared__ __align__(16) int   dec_lds[T_LEN];
    __shared__ unsigned char bp_lds[T_LEN * K_TAGS];            // 16 KB backpointers
    __shared__ unsigned char mask_lds[T_LEN];
    __shared__ int   s_len;
    __shared__ float s_score;

    const int tid = threadIdx.x;
    const int b   = blockIdx.x;
    const float* featb = feats + (size_t)b * T_LEN * K_TAGS;

    if (tid == 0) s_len = 0;

    // ---- Wave 0: issue TDM async load of this batch's 128x128 f32 feature tile.
    if ((tid >> 5) == 0) {
        unsigned long long ga = (unsigned long long)(size_t)featb;
        unsigned lds_addr = (unsigned)(size_t)&feats_lds[0];    // low 32b of flat LDS addr

        v4u g0;
        g0[0] = 1u;                                             // count=1 (valid user D#)
        g0[1] = lds_addr;                                       // lds_addr[31:0]
        g0[2] = (unsigned)(ga & 0xFFFFFFFFull);                 // global_addr[31:0]
        g0[3] = (unsigned)((ga >> 32) & 0x1FFFFFFull)           // global_addr[56:32]
              | (2u << 30);                                     // type=2 ("image")

        v8i g1;
        g1[0] = 0x00020000;                                     // data_size=2 -> 4 bytes
        g1[1] = (int)(128u << 16);                              // tensor_dim0 = 128
        g1[2] = (int)(128u << 16);                              // tensor_dim1 = 128
        g1[3] = (int)(128u << 16);                              // tile_dim0   = 128
        g1[4] = 128;                                            // tile_dim1=128, tile_dim2=0
        g1[5] = 128;                                            // tensor_dim0_stride = 128
        g1[6] = 0;
        g1[7] = 0;

        v4i g2 = {0, 0, 0, 0};
        v4i g3 = {0, 0, 0, 0};
        v8i g4 = {0, 0, 0, 0, 0, 0, 0, 0};
        __builtin_amdgcn_tensor_load_to_lds(g0, g1, g2, g3, g4, 0);
    }

    // ---- Meanwhile: transpose trans[prev][cur] -> transT[cur*PITCH + prev].
    // 32 iterations of coalesced float4 global loads covering all 128x128.
    #pragma unroll 4
    for (int it = 0; it < 32; ++it) {
        int base = it * 512 + tid * 4;      // flat element index, 16B aligned
        int r  = base >> 7;                 // prev (row)
        int c0 = base & 127;                // cur  (col)
        float4 v = *(const float4*)(trans + base);
        transT[(c0 + 0) * PITCH + r] = v.x;
        transT[(c0 + 1) * PITCH + r] = v.y;
        transT[(c0 + 2) * PITCH + r] = v.z;
        transT[(c0 + 3) * PITCH + r] = v.w;
    }
    __syncthreads();   // s_len zeroed, transT writes issued

    // ---- Sequence length (mask is a monotone prefix): wave ballot + LDS atomic.
    unsigned char m = masks[(size_t)b * T_LEN + tid];
    mask_lds[tid] = m;
    unsigned long long bal = __ballot(m != 0);
    if ((tid & 31) == 0) atomicAdd(&s_len, (int)__popcll(bal));

    // Wave 0 waits for the TDM transfer before anyone reads feats_lds.
    if ((tid >> 5) == 0) __builtin_amdgcn_s_wait_tensorcnt(0);
    __syncthreads();

    const int last_idx = s_len - 1;
    const int c = tid;

    // ---- t = 0: partition = feats[:,0,:] + trans[START,:]  (into buffer 0)
    {
        float p0 = feats_lds[c] + transT[c * PITCH + START_TAG];
        part_ab[0][c] = p0;
        if (last_idx == 0) part_save[c] = p0;
    }
    __syncthreads();

    // ---- Forward recurrence: max-plus matvec per timestep, all in LDS.
    // Step t reads part_ab[(t+1)&1], writes part_ab[t&1]: one barrier per step.
    const float* trow = &transT[c * PITCH];
    for (int t = 1; t < T_LEN; ++t) {
        const float* pin = part_ab[(t + 1) & 1];
        float best = -3.402823466e38f;
        int bpi = 0;
        #pragma unroll 8
        for (int p4 = 0; p4 < K_TAGS; p4 += 4) {
            float4 pv = *(const float4*)&pin[p4];        // broadcast ds_load_b128
            float4 tv = *(const float4*)&trow[p4];       // conflict-free ds_load_b128
            float s0 = pv.x + tv.x; if (s0 > best) { best = s0; bpi = p4;     }
            float s1 = pv.y + tv.y; if (s1 > best) { best = s1; bpi = p4 + 1; }
            float s2 = pv.z + tv.z; if (s2 > best) { best = s2; bpi = p4 + 2; }
            float s3 = pv.w + tv.w; if (s3 > best) { best = s3; bpi = p4 + 3; }
        }
        float np = best + feats_lds[t * K_TAGS + c];
        part_ab[t & 1][c] = np;
        bp_lds[t * K_TAGS + c] = (unsigned char)(mask_lds[t] ? bpi : 0);
        if (t == last_idx) part_save[c] = np;
        __syncthreads();
    }

    // ---- Transition to STOP from partition at last valid position.
    red_f[c] = part_save[c] + transT[STOP_TAG * PITCH + c];   // thread c = prev c
    __syncthreads();

    if (tid == 0) {
        float bestv = -3.402823466e38f;
        int ptr = 0;
        for (int p = 0; p < K_TAGS; ++p) {               // first-max tie-break
            float v = red_f[p];
            if (v > bestv) { bestv = v; ptr = p; }
        }
        s_score = bestv;

        // ---- Backtrace (replicates reference's masked-bp + overwritten-row logic).
        dec_lds[T_LEN - 1] = ptr;
        int cur = ptr;
        for (int j = T_LEN - 2; j >= 0; --j) {
            int np = (j == last_idx) ? ptr
                                     : (int)bp_lds[(j + 1) * K_TAGS + cur];
            dec_lds[j] = np;
            cur = np;
        }
    }
    __syncthreads();

    out_idx[(size_t)b * T_LEN + tid] = dec_lds[tid];     // coalesced row write
    if (tid == 0) out_score[b] = s_score;
}

extern "C" void kernel_launch(void* const* d_in, const int* in_sizes, int n_in,
                              void* d_out, int out_size, void* d_ws, size_t ws_size,
                              hipStream_t stream) {
    const float*         feats = (const float*)d_in[0];
    const float*         trans = (const float*)d_in[1];
    const unsigned char* masks = (const unsigned char*)d_in[2];   // jnp bool -> 1 byte

    const int B = in_sizes[0] / (T_LEN * K_TAGS);

    float* out_score = (float*)d_out;          // (B,) float32
    int*   out_idx   = (int*)d_out + B;        // (B,T) int32, concatenated after scores

    viterbi_crf_kernel<<<B, 128, 0, stream>>>(feats, trans, masks, out_score, out_idx);
}